// DyGraphConv_1632087572829
// MI455X (gfx1250) — compile-verified
//
#include <hip/hip_runtime.h>

typedef __attribute__((ext_vector_type(2))) float v2f;
typedef __attribute__((ext_vector_type(8))) float v8f;

#define BATCH 2
#define CH    192
#define NPTS  4096
#define KNN   9
#define COUT  192
#define KDIM  384   // 2*CH

// ------------- Kernel 1: L2-normalize over channels, transpose to (B,N,C) -------------
__global__ __launch_bounds__(256) void k_normalize(const float* __restrict__ x,
                                                   float* __restrict__ xn) {
  const int tx = threadIdx.x;            // point within group of 32 (coalesced dim)
  const int ty = threadIdx.y;            // channel slice 0..7
  const int n  = blockIdx.x * 32 + tx;
  const int b  = blockIdx.y;
  const float* xb = x + (size_t)b * CH * NPTS;

  float v[24];
  float ss = 0.f;
#pragma unroll
  for (int i = 0; i < 24; ++i) {
    int c = ty + i * 8;
    float t = xb[(size_t)c * NPTS + n];
    v[i] = t;
    ss += t * t;
  }
  __shared__ float red[8][32];
  __shared__ float invn[32];
  red[ty][tx] = ss;
  __syncthreads();
  if (ty == 0) {
    float s = 0.f;
#pragma unroll
    for (int j = 0; j < 8; ++j) s += red[j][tx];
    float nrm = fmaxf(sqrtf(s), 1e-12f);
    invn[tx] = 1.0f / nrm;
  }
  __syncthreads();
  const float inv = invn[tx];
  float* outp = xn + ((size_t)(b * NPTS + n)) * CH;
#pragma unroll
  for (int i = 0; i < 24; ++i) outp[ty + i * 8] = v[i] * inv;
}

// ------------- Kernel 2: fused Gram (WMMA f32) + top-9 neighbor selection -------------
// block = 256 threads (8 waves); block owns 16 query rows; each wave sweeps 32 col tiles.
__global__ __launch_bounds__(256) void k_knn(const float* __restrict__ xn,
                                             int* __restrict__ nn_idx) {
  const int tid  = threadIdx.x;
  const int wave = tid >> 5;
  const int lane = tid & 31;
  const int b    = blockIdx.y;
  const int i0   = blockIdx.x * 16;

  __shared__ float Arows[16][CH];          // 12 KB: the 16 query rows
  __shared__ float tileBuf[8][16][16];     // 8 KB: per-wave tile spill for transpose
  __shared__ float mVal[8][16][KNN];       // per-wave top-9 partials
  __shared__ int   mIdx[8][16][KNN];

  const float* xnb = xn + (size_t)b * NPTS * CH;

  for (int e = tid; e < 16 * CH; e += 256) {
    int r = e / CH, c = e % CH;
    Arows[r][c] = xnb[(size_t)(i0 + r) * CH + c];
  }
  __syncthreads();

  const int m16  = lane & 15;
  const int koff = (lane >> 4) * 2;

  float tv[KNN]; int ti[KNN];
#pragma unroll
  for (int s = 0; s < KNN; ++s) { tv[s] = -3.4e38f; ti[s] = 0x7fffffff; }

  for (int step = 0; step < 32; ++step) {          // uniform trip count for barriers
    const int jt = wave + step * 8;
    const int j0 = jt * 16;
    const float* Brow = xnb + (size_t)(j0 + m16) * CH + koff;

    v8f acc = {0.f,0.f,0.f,0.f,0.f,0.f,0.f,0.f};
#pragma unroll 4
    for (int k = 0; k < CH; k += 4) {
      v2f a  = *(const v2f*)&Arows[m16][k + koff]; // A[m][k0], A[m][k0+1]
      v2f bb = *(const v2f*)(Brow + k);            // B[k0][n],  B[k0+1][n]
      acc = __builtin_amdgcn_wmma_f32_16x16x4_f32(false, a, false, bb,
                                                  (short)0, acc, false, false);
    }
    // spill tile: acc[r] on lane l is Gram[i0 + r + 8*(l>>4)][j0 + (l&15)]
    const int rbase = (lane >> 4) * 8;
#pragma unroll
    for (int r = 0; r < 8; ++r) tileBuf[wave][rbase + r][m16] = acc[r];
    __syncthreads();
    if (lane < 16) {                     // lane owns row `lane`; stable top-9 insert
#pragma unroll 1
      for (int col = 0; col < 16; ++col) {
        float v  = tileBuf[wave][lane][col];
        int   id = j0 + col;
        if (v > tv[KNN - 1] || (v == tv[KNN - 1] && id < ti[KNN - 1])) {
          float nv = v; int ni = id;
#pragma unroll
          for (int p = 0; p < KNN; ++p) {
            bool gt = (nv > tv[p]) || (nv == tv[p] && ni < ti[p]);
            float ov = tv[p]; int oi = ti[p];
            tv[p] = gt ? nv : tv[p];  ti[p] = gt ? ni : ti[p];
            nv    = gt ? ov : nv;     ni    = gt ? oi : ni;
          }
        }
      }
    }
    __syncthreads();
  }

  if (lane < 16) {
#pragma unroll
    for (int s = 0; s < KNN; ++s) { mVal[wave][lane][s] = tv[s]; mIdx[wave][lane][s] = ti[s]; }
  }
  __syncthreads();

  if (tid < 16) {                       // merge 8 wave-partials per row
    float fv[KNN]; int fi[KNN];
#pragma unroll
    for (int s = 0; s < KNN; ++s) { fv[s] = -3.4e38f; fi[s] = 0x7fffffff; }
    for (int w = 0; w < 8; ++w) {
      for (int s = 0; s < KNN; ++s) {
        float nv = mVal[w][tid][s]; int ni = mIdx[w][tid][s];
        if (nv > fv[KNN - 1] || (nv == fv[KNN - 1] && ni < fi[KNN - 1])) {
#pragma unroll
          for (int p = 0; p < KNN; ++p) {
            bool gt = (nv > fv[p]) || (nv == fv[p] && ni < fi[p]);
            float ov = fv[p]; int oi = fi[p];
            fv[p] = gt ? nv : fv[p];  fi[p] = gt ? ni : fi[p];
            nv    = gt ? ov : nv;     ni    = gt ? oi : ni;
          }
        }
      }
    }
    int* dst = nn_idx + ((size_t)(b * NPTS + i0 + tid)) * KNN;
#pragma unroll
    for (int s = 0; s < KNN; ++s) dst[s] = fi[s];
  }
}

// ------------- Kernel 3: max-relative aggregation  x_rel = max_k(x_j - x_i) -------------
__global__ __launch_bounds__(256) void k_maxrel(const float* __restrict__ x,
                                                const int* __restrict__ nn_idx,
                                                float* __restrict__ xrel) {
  int id = blockIdx.x * 256 + threadIdx.x;
  if (id >= BATCH * CH * NPTS) return;
  int n  = id & (NPTS - 1);
  int bc = id >> 12;
  int c  = bc % CH;
  int b  = bc / CH;
  const float* xc  = x + (size_t)(b * CH + c) * NPTS;
  const int*   idx = nn_idx + (size_t)(b * NPTS + n) * KNN;
  float center = xc[n];
  float m = -3.4e38f;
#pragma unroll
  for (int k = 0; k < KNN; ++k) m = fmaxf(m, xc[idx[k]] - center);
  xrel[id] = m;
}

// ------------- Kernel 4: 1x1 conv GEMM (WMMA f32) + bias + ReLU ---------------------
// one wave per 16x16 tile; feat is read interleaved: even k -> x, odd k -> x_rel
__global__ __launch_bounds__(32) void k_gemm(const float* __restrict__ Wm,
                                             const float* __restrict__ bias,
                                             const float* __restrict__ x,
                                             const float* __restrict__ xrel,
                                             float* __restrict__ out) {
  const int lane  = threadIdx.x;
  const int ntile = blockIdx.x, mtile = blockIdx.y, b = blockIdx.z;
  const int m16   = lane & 15;
  const int hi    = lane >> 4;
  const int koff  = hi * 2;
  const int n     = ntile * 16 + m16;

  const float* wrow = Wm + (size_t)(mtile * 16 + m16) * KDIM + koff;
  const float* xb   = x    + (size_t)b * CH * NPTS + n;
  const float* rb   = xrel + (size_t)b * CH * NPTS + n;

  v8f acc = {0.f,0.f,0.f,0.f,0.f,0.f,0.f,0.f};
#pragma unroll 4
  for (int t = 0; t < KDIM / 4; ++t) {
    int c = 2 * t + hi;                       // channel = (4t + koff)/2, always integral
    v2f a = *(const v2f*)(wrow + 4 * t);      // W[m][k0], W[m][k0+1]
    v2f bb;
    bb.x = xb[(size_t)c * NPTS];              // feat[k0][n]   (even -> x)
    bb.y = rb[(size_t)c * NPTS];              // feat[k0+1][n] (odd  -> x_rel)
    acc = __builtin_amdgcn_wmma_f32_16x16x4_f32(false, a, false, bb,
                                                (short)0, acc, false, false);
  }
#pragma unroll
  for (int r = 0; r < 8; ++r) {
    int m = mtile * 16 + r + 8 * hi;
    float v = fmaxf(acc[r] + bias[m], 0.f);
    out[(size_t)(b * COUT + m) * NPTS + n] = v;
  }
}

extern "C" void kernel_launch(void* const* d_in, const int* in_sizes, int n_in,
                              void* d_out, int out_size, void* d_ws, size_t ws_size,
                              hipStream_t stream) {
  const float* x    = (const float*)d_in[0];   // (B, C, 16,16,16) = (B, C, N)
  const float* Wm   = (const float*)d_in[1];   // (Cout, 2C)
  const float* bias = (const float*)d_in[2];   // (Cout,)
  float* out = (float*)d_out;                  // (B, Cout, N)

  char* ws = (char*)d_ws;
  float* xn   = (float*)ws;                                          // B*N*C f32
  float* xrel = (float*)(ws + (size_t)BATCH * NPTS * CH * 4);        // B*C*N f32
  int*   nn   = (int*)(ws + (size_t)2 * BATCH * NPTS * CH * 4);      // B*N*K int

  k_normalize<<<dim3(NPTS / 32, BATCH), dim3(32, 8), 0, stream>>>(x, xn);
  k_knn<<<dim3(NPTS / 16, BATCH), 256, 0, stream>>>(xn, nn);
  k_maxrel<<<dim3((BATCH * CH * NPTS) / 256), 256, 0, stream>>>(x, nn, xrel);
  k_gemm<<<dim3(NPTS / 16, COUT / 16, BATCH), 32, 0, stream>>>(Wm, bias, x, xrel, out);
}